// SAConv_67817533604421
// MI455X (gfx1250) — compile-verified
//
#include <hip/hip_runtime.h>
#include <hip/hip_bf16.h>

typedef __attribute__((ext_vector_type(2))) float v2f;
typedef __attribute__((ext_vector_type(8))) float v8f;

#define GRP    64
#define HW     56
#define KWIN   7
#define KK     49     // 7*7
#define TILE_H 8
#define LDS_W  62     // 56 + 2*3
#define LDS_H  14     // TILE_H + 2*3

// ---------------------------------------------------------------------------
// Kernel 1: r_o[g][kk] = sum_{c<4} Wr[g*4+c] * r[(g*4+c)*49 + kk]
// Block-diagonal GEMM (64x49 = [64x256-blockdiag] x [256x49]) on
// V_WMMA_F32_16X16X4_F32.  16 blocks (4 M-tiles x 4 N-tiles), 1 wave each,
// 16 accumulating WMMAs per wave.
// A-frag (16x4 f32): lanes 0-15 hold {K0,K1}, lanes 16-31 hold {K2,K3};
// at K-step t only row t is nonzero (the 4 Wr weights of group 16m+t).
// B-frag (4x16 f32): mirrored layout. B loads use a CLAMPED column index
// (no branch): B column n only affects D column n, and columns >= 49 are
// never stored, so garbage there is dead.  C/D: lane&15 = N, lane>>4 picks
// M rows 0-7 / 8-15 across the 8 accumulator VGPRs.
// ---------------------------------------------------------------------------
__global__ __launch_bounds__(32)
void saconv_ro_wmma(const float* __restrict__ Wr,
                    const float* __restrict__ rwin,
                    float* __restrict__ ro) {
  const int m    = blockIdx.x & 3;   // group tile: g in [16m, 16m+16)
  const int n    = blockIdx.x >> 2;  // kk tile:    kk in [16n, 16n+16)
  const int lane = threadIdx.x & 31;
  const int row  = lane & 15;
  const int hi   = lane >> 4;        // 0: K=0,1   1: K=2,3
  const int col  = 16 * n + row;     // kk column for B / D
  const int colc = (col < KK) ? col : (KK - 1);  // clamped, branch-free loads

  v8f acc = {0.f, 0.f, 0.f, 0.f, 0.f, 0.f, 0.f, 0.f};

  for (int t = 0; t < 16; ++t) {
    const int g = 16 * m + t;
    const float w0 = Wr[4 * g + 2 * hi + 0];
    const float w1 = Wr[4 * g + 2 * hi + 1];
    v2f a;
    a.x = (row == t) ? w0 : 0.0f;
    a.y = (row == t) ? w1 : 0.0f;
    v2f b;
    b.x = rwin[(4 * g + 2 * hi + 0) * KK + colc];
    b.y = rwin[(4 * g + 2 * hi + 1) * KK + colc];
    acc = __builtin_amdgcn_wmma_f32_16x16x4_f32(
        /*neg_a=*/false, a, /*neg_b=*/false, b,
        /*c_mod=*/(short)0, acc, /*reuse_a=*/false, /*reuse_b=*/false);
  }

  if (col < KK) {
#pragma unroll
    for (int j = 0; j < 8; ++j) {
      const int gOut = 16 * m + hi * 8 + j;   // C/D layout: M = hi*8 + vgpr
      ro[gOut * KK + col] = acc[j];
    }
  }
}

// ---------------------------------------------------------------------------
// Kernel 2: sliding-window softmax attention, D=1 -> pure per-channel scalars.
// One block = one (b,g) plane x one 8-row stripe. Stage 14x62 zero-padded
// stripe of x into LDS; each thread owns pixels, keeps 49 logits in VGPRs,
// native v_exp_f32 softmax, weighted sum, one global store.
// ---------------------------------------------------------------------------
__global__ __launch_bounds__(256)
void saconv_attn(const float* __restrict__ x,
                 const float* __restrict__ Wq,
                 const float* __restrict__ Wk,
                 const float* __restrict__ Wv,
                 const float* __restrict__ up,
                 const float* __restrict__ vp,
                 const float* __restrict__ ro,
                 float* __restrict__ out) {
  __shared__ float xs[LDS_H * LDS_W];
  __shared__ float ros[KK];

  const int tid  = threadIdx.x;
  const int g    = blockIdx.x & (GRP - 1);
  const int bIdx = blockIdx.x >> 6;
  const int r0   = blockIdx.y * TILE_H;
  const int base = (bIdx * GRP + g) * HW * HW;

  if (tid < KK) ros[tid] = ro[g * KK + tid];

  for (int i = tid; i < LDS_H * LDS_W; i += 256) {
    const int rr  = i / LDS_W;
    const int cc  = i - rr * LDS_W;
    const int row = r0 - 3 + rr;
    const int col = cc - 3;
    float v = 0.0f;
    if ((unsigned)row < (unsigned)HW && (unsigned)col < (unsigned)HW)
      v = x[base + row * HW + col];
    xs[i] = v;
  }
  __syncthreads();

  const float wq = Wq[g], wk = Wk[g], wv = Wv[g];
  const float uu = up[g], vv = vp[g];

  for (int p = tid; p < TILE_H * HW; p += 256) {
    const int py = p / HW;
    const int px = p - py * HW;
    const float* w0 = &xs[py * LDS_W + px];

    const float center = w0[3 * LDS_W + 3];
    const float q  = wq * center;
    const float a  = (q + uu) * wk;   // multiplies raw x (k_w = wk * window)
    const float qv = q + vv;

    float l[KK];
    float mx = -3.0e38f;
#pragma unroll
    for (int i = 0; i < KWIN; ++i)
#pragma unroll
      for (int j = 0; j < KWIN; ++j) {
        const int kk = i * KWIN + j;
        const float t = fmaf(a, w0[i * LDS_W + j], qv * ros[kk]);
        l[kk] = t;
        mx = fmaxf(mx, t);
      }

    float s = 0.0f, o = 0.0f;
#pragma unroll
    for (int i = 0; i < KWIN; ++i)
#pragma unroll
      for (int j = 0; j < KWIN; ++j) {
        const int kk = i * KWIN + j;
        const float e = __expf(l[kk] - mx);
        s += e;
        o = fmaf(e, w0[i * LDS_W + j], o);
      }

    out[base + (r0 + py) * HW + px] = wv * o / s;
  }
}

extern "C" void kernel_launch(void* const* d_in, const int* in_sizes, int n_in,
                              void* d_out, int out_size, void* d_ws, size_t ws_size,
                              hipStream_t stream) {
  const float* x  = (const float*)d_in[0];
  const float* r  = (const float*)d_in[1];
  const float* Wq = (const float*)d_in[2];
  const float* Wk = (const float*)d_in[3];
  const float* Wv = (const float*)d_in[4];
  const float* Wr = (const float*)d_in[5];
  const float* up = (const float*)d_in[6];
  const float* vp = (const float*)d_in[7];
  float* out = (float*)d_out;
  float* ro  = (float*)d_ws;   // 64*49 floats

  // Stage 1: r_o precompute on WMMA (16 waves, 4x4 tiles of the 64x49 output).
  saconv_ro_wmma<<<dim3(16), dim3(32), 0, stream>>>(Wr, r, ro);

  // Stage 2: attention. B*G = 4*64 = 256 planes, 7 stripes of 8 rows each.
  saconv_attn<<<dim3(256, 7), dim3(256), 0, stream>>>(x, Wq, Wk, Wv, up, vp, ro, out);
}